// LSTMNet_13340168421446
// MI455X (gfx1250) — compile-verified
//
#include <hip/hip_runtime.h>

#define T_STEPS 8192
#define HSZ 10
#define HPAD 12   // pad K to 12 so 16x16x4 WMMA K-chunks are b64-aligned

typedef __attribute__((ext_vector_type(2))) float v2f;
typedef __attribute__((ext_vector_type(8))) float v8f;

// Fast activations: one v_exp_f32 + one v_rcp_f32 each (critical path x8192).
__device__ __forceinline__ float fexp(float x) {
    return __builtin_amdgcn_exp2f(x * 1.44269504088896340736f);
}
__device__ __forceinline__ float fsigm(float x) {
    return __builtin_amdgcn_rcpf(1.0f + fexp(-x));
}
__device__ __forceinline__ float ftanh(float x) {
    return 2.0f * fsigm(2.0f * x) - 1.0f;
}

// ---------------------------------------------------------------------------
// Kernel 1: the serial recurrence. One wave; lane j (j<10) owns hidden unit j.
// ---------------------------------------------------------------------------
__global__ void lstm_seq_kernel(const float* __restrict__ X,
                                const float* __restrict__ W_ih,
                                const float* __restrict__ W_hh,
                                const float* __restrict__ b_ih,
                                const float* __restrict__ b_hh,
                                const float* __restrict__ W_lin,
                                const float* __restrict__ b_lin,
                                const float* __restrict__ h0,
                                const float* __restrict__ c0,
                                float* __restrict__ Hbuf,   // [T, HPAD] scratch (split path)
                                float* __restrict__ out,    // [T] (fused path)
                                int fuse)
{
    const int j  = threadIdx.x;            // lane id, wave32
    const int jj = (j < HSZ) ? j : 0;      // lanes >= 10 mirror lane 0 (kept finite)

    // One-time weight load into VGPRs: rows j, j+10, j+20, j+30 of W_hh/W_ih/bias.
    float wi[4], bb[4], whh[4][HSZ];
    #pragma unroll
    for (int g = 0; g < 4; ++g) {
        const int row = g * HSZ + jj;
        wi[g] = W_ih[row];                 // D == 1
        bb[g] = b_ih[row] + b_hh[row];
        #pragma unroll
        for (int k = 0; k < HSZ; ++k) whh[g][k] = W_hh[row * HSZ + k];
    }
    float wl[HSZ];
    #pragma unroll
    for (int k = 0; k < HSZ; ++k) wl[k] = W_lin[k];
    const float bl = b_lin[0];

    float h[HSZ];
    #pragma unroll
    for (int k = 0; k < HSZ; ++k) h[k] = h0[k];
    float c = c0[jj];

    for (int t = 0; t < T_STEPS; ++t) {
        if (t + 128 < T_STEPS)
            __builtin_prefetch(X + t + 128, 0, 0);   // global_prefetch_b8
        const float x = X[t];

        float a0 = bb[0] + x * wi[0];
        float a1 = bb[1] + x * wi[1];
        float a2 = bb[2] + x * wi[2];
        float a3 = bb[3] + x * wi[3];
        #pragma unroll
        for (int k = 0; k < HSZ; ++k) {    // 4 independent FMA chains of depth 10
            a0 += h[k] * whh[0][k];
            a1 += h[k] * whh[1][k];
            a2 += h[k] * whh[2][k];
            a3 += h[k] * whh[3][k];
        }

        const float ig = fsigm(a0);
        const float fg = fsigm(a1);
        const float gg = ftanh(a2);
        const float og = fsigm(a3);
        c = fg * c + ig * gg;              // c_j stays in lane j forever
        const float hn = og * ftanh(c);

        // Broadcast h_new to every lane (10 wave32 shuffles).
        #pragma unroll
        for (int k = 0; k < HSZ; ++k) h[k] = __shfl(hn, k, 32);

        if (fuse) {
            float o = bl;
            #pragma unroll
            for (int k = 0; k < HSZ; ++k) o += h[k] * wl[k];
            if (j == 0) out[t] = o;
        } else {
            // 48B coalesced store of the padded h row; pad cols zeroed.
            if (j < HPAD) Hbuf[t * HPAD + j] = (j < HSZ) ? hn : 0.0f;
        }
    }
}

// ---------------------------------------------------------------------------
// Kernel 2: out[T] = H[T,10] @ W_lin^T + b_lin via V_WMMA_F32_16X16X4_F32.
// One wave per 16-row tile; 3 chained WMMAs cover K=12 (10 real + 2 zero-pad).
// W_lin is replicated across all 16 B columns, so every C column equals the
// result vector; lane 0 / lane 16 extract rows 0-7 / 8-15 (documented layout).
// ---------------------------------------------------------------------------
__global__ void proj_wmma_kernel(const float* __restrict__ Hbuf,
                                 const float* __restrict__ W_lin,
                                 const float* __restrict__ b_lin,
                                 float* __restrict__ out)
{
    const int l    = threadIdx.x;
    const int row0 = blockIdx.x * 16;
    const int m    = l & 15;          // A-matrix row for this lane
    const int kh   = (l >> 4) * 2;    // lanes 0-15 hold K+0,K+1 ; lanes 16-31 hold K+2,K+3

    v8f acc = {};
    #pragma unroll
    for (int ck = 0; ck < 3; ++ck) {
        const int kb = ck * 4 + kh;
        v2f a, b;
        a.x = Hbuf[(row0 + m) * HPAD + kb];
        a.y = Hbuf[(row0 + m) * HPAD + kb + 1];
        b.x = (kb     < HSZ) ? W_lin[kb]     : 0.0f;   // same value in all N columns
        b.y = (kb + 1 < HSZ) ? W_lin[kb + 1] : 0.0f;
        acc = __builtin_amdgcn_wmma_f32_16x16x4_f32(
                  /*neg_a=*/false, a, /*neg_b=*/false, b,
                  /*c_mod=*/(short)0, acc, /*reuse_a=*/false, /*reuse_b=*/false);
    }

    const float bl = b_lin[0];
    if (l == 0) {
        #pragma unroll
        for (int r = 0; r < 8; ++r) out[row0 + r] = acc[r] + bl;
    } else if (l == 16) {
        #pragma unroll
        for (int r = 0; r < 8; ++r) out[row0 + 8 + r] = acc[r] + bl;
    }
}

extern "C" void kernel_launch(void* const* d_in, const int* in_sizes, int n_in,
                              void* d_out, int out_size, void* d_ws, size_t ws_size,
                              hipStream_t stream)
{
    (void)in_sizes; (void)n_in; (void)out_size;
    const float* X     = (const float*)d_in[0];
    const float* W_ih  = (const float*)d_in[1];
    const float* W_hh  = (const float*)d_in[2];
    const float* b_ih  = (const float*)d_in[3];
    const float* b_hh  = (const float*)d_in[4];
    const float* W_lin = (const float*)d_in[5];
    const float* b_lin = (const float*)d_in[6];
    const float* h0    = (const float*)d_in[7];
    const float* c0    = (const float*)d_in[8];
    float* out = (float*)d_out;

    const size_t need = (size_t)T_STEPS * HPAD * sizeof(float);
    if (ws_size >= need) {
        float* Hbuf = (float*)d_ws;
        lstm_seq_kernel<<<1, 32, 0, stream>>>(X, W_ih, W_hh, b_ih, b_hh,
                                              W_lin, b_lin, h0, c0,
                                              Hbuf, out, /*fuse=*/0);
        proj_wmma_kernel<<<T_STEPS / 16, 32, 0, stream>>>(Hbuf, W_lin, b_lin, out);
    } else {
        lstm_seq_kernel<<<1, 32, 0, stream>>>(X, W_ih, W_hh, b_ih, b_hh,
                                              W_lin, b_lin, h0, c0,
                                              nullptr, out, /*fuse=*/1);
    }
}